// Sampler_51419348468365
// MI455X (gfx1250) — compile-verified
//
#include <hip/hip_runtime.h>
#include <stdint.h>

// ---------------- problem constants (match reference) ----------------
#define N_CANDI 8388608
#define N_SAMPI 1048576
#define N_SEGI  4096

#define BLK  256           // 8 wave32 waves per block
#define VPT  16
#define ITEMS (BLK * VPT)  // 4096 candidates per block
#define NBLK1 (N_CANDI / ITEMS)

// Order-preserving encoding of float into unsigned so unsigned max == float max.
// All real floats encode to > 0, so 0 is a safe identity for max.
__device__ __forceinline__ unsigned enc_f32(float f) {
  unsigned u = __float_as_uint(f);
  return (u & 0x80000000u) ? ~u : (u | 0x80000000u);
}
__device__ __forceinline__ float dec_f32(unsigned e) {
  return (e & 0x80000000u) ? __uint_as_float(e ^ 0x80000000u)
                           : __uint_as_float(~e);
}

// ---------------- gfx1250 async global->LDS helpers ----------------
typedef __attribute__((address_space(1))) int*  as1_i32_p;   // prints as "__device__ int *"
typedef __attribute__((address_space(3))) void* as3_void_p;  // LDS
typedef __attribute__((address_space(3))) int*  as3_i32_p;

__device__ __forceinline__ void async_ld_b32(const float* gsrc, float* lds_dst) {
#if __has_builtin(__builtin_amdgcn_global_load_async_to_lds_b32)
  as1_i32_p g = (as1_i32_p)(uintptr_t)gsrc;          // int -> AS1 pointer (same repr)
  as3_i32_p l = (as3_i32_p)(as3_void_p)(void*)lds_dst; // generic -> AS3, then bitcast
  __builtin_amdgcn_global_load_async_to_lds_b32(g, l, 0, 0);
#else
  unsigned lo = (unsigned)(uintptr_t)(as3_void_p)(void*)lds_dst;
  unsigned long long ga = (unsigned long long)(uintptr_t)gsrc;
  asm volatile("global_load_async_to_lds_b32 %0, %1, off"
               :: "v"(lo), "v"(ga) : "memory");
#endif
}

__device__ __forceinline__ void wait_async_le1() {
#if __has_builtin(__builtin_amdgcn_s_wait_asynccnt)
  __builtin_amdgcn_s_wait_asynccnt(1);
#else
  asm volatile("s_wait_asynccnt 0x1" ::: "memory");
#endif
  asm volatile("" ::: "memory");
}

__device__ __forceinline__ void wait_async_le0() {
#if __has_builtin(__builtin_amdgcn_s_wait_asynccnt)
  __builtin_amdgcn_s_wait_asynccnt(0);
#else
  asm volatile("s_wait_asynccnt 0x0" ::: "memory");
#endif
  asm volatile("" ::: "memory");
}

// ---------------- kernels ----------------
__global__ void sm_init(unsigned* __restrict__ segmax, float* __restrict__ segsum) {
  int t = blockIdx.x * blockDim.x + threadIdx.x;
  if (t < N_SEGI) { segmax[t] = 0u; segsum[t] = 0.0f; }
}

// Pass 1: y = logits[edge_id] + loglog_u ; loglog_u streamed via the gfx1250
// async global->LDS path (double-buffered global_load_async_to_lds_b32 +
// s_wait_asynccnt); per-segment max via LDS ds_max_u32, flushed with one
// global atomic per (block, touched segment).
__global__ void sm_pass1(const float* __restrict__ logits,
                         const int* __restrict__ cand,      // (N_CAND, 5) int32
                         const float* __restrict__ llu,
                         float* __restrict__ y,
                         unsigned short* __restrict__ seg16,
                         unsigned* __restrict__ segmax) {
  __shared__ unsigned smax[N_SEGI];            // 16 KB of 320 KB WGP LDS
  __shared__ float lbuf[2][BLK];               // 2 KB async staging buffers
  for (int s = threadIdx.x; s < N_SEGI; s += BLK) smax[s] = 0u;
  __syncthreads();

  const int tid  = (int)threadIdx.x;
  const int base = blockIdx.x * ITEMS;

  // prologue: stage chunk 0
  async_ld_b32(llu + base + tid, &lbuf[0][tid]);

#pragma unroll
  for (int k = 0; k < VPT; ++k) {
    const int p = k & 1;
    if (k + 1 < VPT) {
      // buffer p^1 was consumed last iteration -> safe to overwrite
      async_ld_b32(llu + base + (k + 1) * BLK + tid, &lbuf[p ^ 1][tid]);
      wait_async_le1();                        // chunk k done (in-order), k+1 in flight
    } else {
      wait_async_le0();
    }

    int i = base + k * BLK + tid;
    __builtin_prefetch(cand + (size_t)(i + ITEMS) * 5, 0, 1); // global_prefetch_b8
    int seg = cand[i * 5 + 0];
    int eid = cand[i * 5 + 1];
    float v = logits[eid] + lbuf[p][tid];      // gather + staged stream
    y[i] = v;
    seg16[i] = (unsigned short)seg;
    atomicMax(&smax[seg], enc_f32(v));         // ds_max_u32
  }
  __syncthreads();

  // segments are sorted: block touches contiguous range [s0, s1]
  int s0 = cand[(size_t)base * 5];
  int s1 = cand[(size_t)(base + ITEMS - 1) * 5];
  for (int s = s0 + tid; s <= s1; s += BLK) {
    unsigned v = smax[s];
    if (v) atomicMax(&segmax[s], v);           // global_atomic_max_u32 (few per block)
  }
}

// Pass 2: e = exp(y - m[seg]) in-place; per-segment sum via LDS ds_add_f32,
// flushed with one global_atomic_add_f32 per touched segment.
__global__ void sm_pass2(float* __restrict__ ye,                 // in: y, out: e
                         const unsigned short* __restrict__ seg16,
                         const unsigned* __restrict__ segmax,
                         float* __restrict__ segsum) {
  __shared__ float ssum[N_SEGI];
  for (int s = threadIdx.x; s < N_SEGI; s += BLK) ssum[s] = 0.0f;
  __syncthreads();

  const int base = blockIdx.x * ITEMS;
#pragma unroll
  for (int k = 0; k < VPT; ++k) {
    int i = base + k * BLK + (int)threadIdx.x;
    __builtin_prefetch(ye + (i + ITEMS), 0, 1);
    int seg = (int)seg16[i];
    float m = dec_f32(segmax[seg]);            // 16 KB table, cache resident
    float e = __expf(ye[i] - m);
    ye[i] = e;
    atomicAdd(&ssum[seg], e);                  // ds_add_f32
  }
  __syncthreads();

  int s0 = (int)seg16[base];
  int s1 = (int)seg16[base + ITEMS - 1];
  for (int s = s0 + (int)threadIdx.x; s <= s1; s += BLK) {
    float v = ssum[s];
    if (v != 0.0f) atomicAdd(&segsum[s], v);   // global_atomic_add_f32
  }
}

// Pass 3: gather sampled candidates, normalize, straight-through output.
__global__ void sm_pass3(const int* __restrict__ sampled,        // (N_SAMP, 6) int32
                         const unsigned short* __restrict__ seg16,
                         const float* __restrict__ e,
                         const float* __restrict__ segsum,
                         float* __restrict__ out) {
  int j = blockIdx.x * blockDim.x + threadIdx.x;
  if (j >= N_SAMPI) return;
  int ca = sampled[j * 6 + 5];
  float p = e[ca] / segsum[(int)seg16[ca]];
  out[j] = (1.0f - p) + p;                     // == 1.0f exactly; matches reference
}

// ---------------- host-side launch ----------------
extern "C" void kernel_launch(void* const* d_in, const int* in_sizes, int n_in,
                              void* d_out, int out_size, void* d_ws, size_t ws_size,
                              hipStream_t stream) {
  const float* logits  = (const float*)d_in[0];   // edges_logits (5,000,000)
  const int*   cand    = (const int*)d_in[1];     // candidate_edges (8388608 x 5)
  const float* llu     = (const float*)d_in[2];   // loglog_u (8388608)
  const int*   sampled = (const int*)d_in[3];     // sampled_edges (1048576 x 6)
  float*       out     = (float*)d_out;

  char* ws = (char*)d_ws;
  float*          y      = (float*)ws;                                  // 32 MB
  unsigned short* seg16  = (unsigned short*)(ws + (size_t)N_CANDI * 4); // 16 MB
  unsigned*       segmax = (unsigned*)(ws + (size_t)N_CANDI * 6);       // 16 KB
  float*          segsum = (float*)(ws + (size_t)N_CANDI * 6 + (size_t)N_SEGI * 4);

  sm_init<<<(N_SEGI + 255) / 256, 256, 0, stream>>>(segmax, segsum);
  sm_pass1<<<NBLK1, BLK, 0, stream>>>(logits, cand, llu, y, seg16, segmax);
  sm_pass2<<<NBLK1, BLK, 0, stream>>>(y, seg16, segmax, segsum);
  sm_pass3<<<(N_SAMPI + 255) / 256, 256, 0, stream>>>(sampled, seg16, y, segsum, out);
}